// TransitionDown_54778012893610
// MI455X (gfx1250) — compile-verified
//
#include <hip/hip_runtime.h>
#include <hip/hip_bf16.h>
#include <float.h>

#define N_PTS 50000
#define IN_CH 128
#define OUT_CH 256
#define M_CTR 5000
#define KNN 16
#define BN_EPS 1e-5f

#define FPS_NB 16               // cooperative blocks (each on its own WGP)
#define FPS_PTS (N_PTS / FPS_NB)  // 3125 points per block, LDS-resident
#define FPS_T 256

typedef __attribute__((ext_vector_type(2))) float v2f;
typedef __attribute__((ext_vector_type(8))) float v8f;

// ---------------------------------------------------------------------------
// 0) Per-call scratch init (stats + grid-barrier counter). d_ws is poisoned
//    once and never re-poisoned, so this must run every call.
// ---------------------------------------------------------------------------
__global__ void init_ws_kernel(float* __restrict__ stats, int* __restrict__ cnt) {
  stats[threadIdx.x] = 0.0f;  // 512 entries: sum[256], sumsq[256]
  if (threadIdx.x == 0) *cnt = 0;
}

// ---------------------------------------------------------------------------
// 1) Cooperative FPS: 16 blocks, each holds its pos slice (37.5KB) and min_d
//    slice (12.5KB) fully in LDS (320KB/WGP on CDNA5). Zero global traffic in
//    the inner scan. One device-scope barrier per iteration; candidate slots
//    are double-buffered by iteration parity to avoid write-after-read races.
// ---------------------------------------------------------------------------
__global__ __launch_bounds__(FPS_T) void fps_coop_kernel(const float* __restrict__ pos,
                                                         float* __restrict__ cand,  // 2*16*8 floats
                                                         int* __restrict__ cnt,
                                                         int* __restrict__ out_idx) {
  __shared__ float spx[FPS_PTS];
  __shared__ float spy[FPS_PTS];
  __shared__ float spz[FPS_PTS];
  __shared__ float smind[FPS_PTS];
  __shared__ float s_rd[FPS_T / 32];
  __shared__ int s_ri[FPS_T / 32];
  __shared__ int s_widx;
  __shared__ float s_wx, s_wy, s_wz;

  const int t = threadIdx.x;
  const int bb = blockIdx.x;
  const int base = bb * FPS_PTS;
  const int lane = t & 31;
  const int wv = t >> 5;

  // Stage this block's slice into LDS (coalesced, done once).
  for (int i = t; i < FPS_PTS; i += FPS_T) {
    spx[i] = pos[(size_t)(base + i) * 3 + 0];
    spy[i] = pos[(size_t)(base + i) * 3 + 1];
    spz[i] = pos[(size_t)(base + i) * 3 + 2];
    smind[i] = FLT_MAX;
  }
  float lx = pos[0], ly = pos[1], lz = pos[2];  // seed point (index 0)
  int widx = 0;
  __syncthreads();

  for (int m = 0; m < M_CTR; ++m) {
    if (bb == 0 && t == 0) out_idx[m] = widx;

    // LDS-resident scan: update min_d, find local argmax (tie -> smaller idx).
    float best = -1.0f;
    int bli = 0x7fffffff;
    for (int i = t; i < FPS_PTS; i += FPS_T) {
      float dx = spx[i] - lx;
      float dy = spy[i] - ly;
      float dz = spz[i] - lz;
      float d = dx * dx + dy * dy + dz * dz;
      float md = fminf(smind[i], d);
      smind[i] = md;
      if (md > best) { best = md; bli = i; }
    }
    #pragma unroll
    for (int off = 16; off > 0; off >>= 1) {
      float od = __shfl_down(best, off, 32);
      int oi = __shfl_down(bli, off, 32);
      if (od > best || (od == best && oi < bli)) { best = od; bli = oi; }
    }
    if (lane == 0) { s_rd[wv] = best; s_ri[wv] = bli; }
    __syncthreads();

    if (t == 0) {
      #pragma unroll
      for (int w = 1; w < FPS_T / 32; ++w) {
        if (s_rd[w] > best || (s_rd[w] == best && s_ri[w] < bli)) {
          best = s_rd[w];
          bli = s_ri[w];
        }
      }
      // Publish candidate (dist, global idx, coords from our LDS slice).
      float* c = cand + ((m & 1) * FPS_NB + bb) * 8;
      __hip_atomic_store(&c[0], best, __ATOMIC_RELAXED, __HIP_MEMORY_SCOPE_AGENT);
      __hip_atomic_store(&c[1], __int_as_float(base + bli), __ATOMIC_RELAXED, __HIP_MEMORY_SCOPE_AGENT);
      __hip_atomic_store(&c[2], spx[bli], __ATOMIC_RELAXED, __HIP_MEMORY_SCOPE_AGENT);
      __hip_atomic_store(&c[3], spy[bli], __ATOMIC_RELAXED, __HIP_MEMORY_SCOPE_AGENT);
      __hip_atomic_store(&c[4], spz[bli], __ATOMIC_RELAXED, __HIP_MEMORY_SCOPE_AGENT);

      // Grid barrier: release our stores, acquire everyone else's.
      __hip_atomic_fetch_add(cnt, 1, __ATOMIC_ACQ_REL, __HIP_MEMORY_SCOPE_AGENT);
      while (__hip_atomic_load(cnt, __ATOMIC_ACQUIRE, __HIP_MEMORY_SCOPE_AGENT) <
             FPS_NB * (m + 1)) {
        __builtin_amdgcn_s_sleep(1);
      }

      // Redundant winner selection (every block does it -> no second barrier).
      float wbd = -1.0f;
      int wbi = 0x7fffffff;
      float wx = 0.0f, wy = 0.0f, wz = 0.0f;
      const float* cb = cand + (m & 1) * FPS_NB * 8;
      #pragma unroll
      for (int b2 = 0; b2 < FPS_NB; ++b2) {
        float d = __hip_atomic_load(&cb[b2 * 8 + 0], __ATOMIC_RELAXED, __HIP_MEMORY_SCOPE_AGENT);
        int ix = __float_as_int(
            __hip_atomic_load(&cb[b2 * 8 + 1], __ATOMIC_RELAXED, __HIP_MEMORY_SCOPE_AGENT));
        float x = __hip_atomic_load(&cb[b2 * 8 + 2], __ATOMIC_RELAXED, __HIP_MEMORY_SCOPE_AGENT);
        float y = __hip_atomic_load(&cb[b2 * 8 + 3], __ATOMIC_RELAXED, __HIP_MEMORY_SCOPE_AGENT);
        float z = __hip_atomic_load(&cb[b2 * 8 + 4], __ATOMIC_RELAXED, __HIP_MEMORY_SCOPE_AGENT);
        if (d > wbd || (d == wbd && ix < wbi)) { wbd = d; wbi = ix; wx = x; wy = y; wz = z; }
      }
      s_widx = wbi; s_wx = wx; s_wy = wy; s_wz = wz;
    }
    __syncthreads();
    widx = s_widx;
    lx = s_wx; ly = s_wy; lz = s_wz;
  }
}

// ---------------------------------------------------------------------------
// 2) GEMM h = x@W + b with V_WMMA_F32_16X16X4_F32 (full fp32 precision).
//    One wave per 16x16 output tile; 8 waves/block; K=128 in 32 steps of 4.
// ---------------------------------------------------------------------------
__global__ __launch_bounds__(256) void gemm_kernel(const float* __restrict__ x,
                                                   const float* __restrict__ W,
                                                   const float* __restrict__ bias,
                                                   float* __restrict__ h) {
  const int lane = threadIdx.x & 31;
  const int wave = threadIdx.x >> 5;
  const int row0 = blockIdx.x * 16;                 // 3125 row tiles
  const int col0 = (blockIdx.y * 8 + wave) * 16;    // 16 col tiles
  const int n = lane & 15;
  const int hi = lane >> 4;                         // 0: lanes 0-15, 1: 16-31

  v8f c = {};
  const float* xa = x + (size_t)(row0 + n) * IN_CH + 2 * hi;
  const float* wb = W + col0 + n;

  #pragma unroll 8
  for (int k = 0; k < IN_CH; k += 4) {
    v2f a;
    a.x = xa[k];
    a.y = xa[k + 1];
    v2f bf;
    bf.x = wb[(size_t)(k + 2 * hi) * OUT_CH];
    bf.y = wb[(size_t)(k + 2 * hi + 1) * OUT_CH];
    c = __builtin_amdgcn_wmma_f32_16x16x4_f32(false, a, false, bf,
                                              (short)0, c, false, false);
  }

  const float bv = bias[col0 + n];
  const int mbase = row0 + 8 * hi;
  #pragma unroll
  for (int r = 0; r < 8; ++r) {
    h[(size_t)(mbase + r) * OUT_CH + col0 + n] = c[r] + bv;
  }
}

// ---------------------------------------------------------------------------
// 3) BatchNorm stats / finalize / apply+ReLU.
// ---------------------------------------------------------------------------
__global__ __launch_bounds__(256) void bn_stats_kernel(const float* __restrict__ h,
                                                       float* __restrict__ stats) {
  const int c = threadIdx.x;
  float s = 0.0f, ss = 0.0f;
  for (int r = blockIdx.x; r < N_PTS; r += gridDim.x) {
    float v = h[(size_t)r * OUT_CH + c];
    s += v;
    ss += v * v;
  }
  atomicAdd(&stats[c], s);
  atomicAdd(&stats[OUT_CH + c], ss);
}

__global__ void bn_finalize_kernel(const float* __restrict__ stats,
                                   const float* __restrict__ gamma,
                                   const float* __restrict__ beta,
                                   float* __restrict__ scsh) {
  const int c = threadIdx.x;
  const float inv_n = 1.0f / (float)N_PTS;
  float mean = stats[c] * inv_n;
  float var = stats[OUT_CH + c] * inv_n - mean * mean;  // biased var
  float sc = gamma[c] * rsqrtf(var + BN_EPS);
  scsh[c] = sc;
  scsh[OUT_CH + c] = beta[c] - mean * sc;
}

__global__ __launch_bounds__(256) void bn_relu_kernel(float* __restrict__ h,
                                                      const float* __restrict__ scsh) {
  const int c = threadIdx.x;  // blockDim == OUT_CH
  const size_t i = (size_t)blockIdx.x * OUT_CH + c;
  float v = h[i] * scsh[c] + scsh[OUT_CH + c];
  h[i] = fmaxf(v, 0.0f);
}

// ---------------------------------------------------------------------------
// 4) KNN: one thread per centroid, register-resident top-16 (replace worst,
//    rescan; fully unrolled). pos reads are block-uniform -> scalarized /
//    L0-broadcast; prefetch emits global_prefetch_b8.
// ---------------------------------------------------------------------------
__global__ __launch_bounds__(256) void knn_kernel(const float* __restrict__ pos,
                                                  const int* __restrict__ fps_idx,
                                                  const int* __restrict__ batch,
                                                  int* __restrict__ nn_idx,
                                                  float* __restrict__ sub_pos,
                                                  int* __restrict__ sub_batch) {
  const int m = blockIdx.x * blockDim.x + threadIdx.x;
  if (m >= M_CTR) return;
  const int ci = fps_idx[m];
  const float qx = pos[ci * 3 + 0];
  const float qy = pos[ci * 3 + 1];
  const float qz = pos[ci * 3 + 2];
  sub_pos[m * 3 + 0] = qx;
  sub_pos[m * 3 + 1] = qy;
  sub_pos[m * 3 + 2] = qz;
  sub_batch[m] = batch[ci];

  float bd[KNN];
  int bi[KNN];
  #pragma unroll
  for (int j = 0; j < KNN; ++j) { bd[j] = FLT_MAX; bi[j] = -1; }
  float worst = FLT_MAX;

  for (int i = 0; i < N_PTS; ++i) {
    __builtin_prefetch(&pos[(i + 64) * 3], 0, 0);
    float dx = pos[i * 3 + 0] - qx;
    float dy = pos[i * 3 + 1] - qy;
    float dz = pos[i * 3 + 2] - qz;
    float d = dx * dx + dy * dy + dz * dz;
    if (d < worst) {
      bool done = false;
      #pragma unroll
      for (int j = 0; j < KNN; ++j) {
        bool repl = (!done) && (bd[j] == worst);
        if (repl) { bd[j] = d; bi[j] = i; done = true; }
      }
      float w = bd[0];
      #pragma unroll
      for (int j = 1; j < KNN; ++j) w = fmaxf(w, bd[j]);
      worst = w;
    }
  }
  #pragma unroll
  for (int j = 0; j < KNN; ++j) nn_idx[m * KNN + j] = bi[j];
}

// ---------------------------------------------------------------------------
// 5) Gather-max over exactly K neighbors: block per centroid, channel per lane.
// ---------------------------------------------------------------------------
__global__ __launch_bounds__(256) void gather_max_kernel(const float* __restrict__ h,
                                                         const int* __restrict__ nn_idx,
                                                         float* __restrict__ out) {
  const int m = blockIdx.x;
  const int c = threadIdx.x;
  float v = -FLT_MAX;
  #pragma unroll
  for (int k = 0; k < KNN; ++k) {
    int r = nn_idx[m * KNN + k];
    v = fmaxf(v, h[(size_t)r * OUT_CH + c]);
  }
  out[(size_t)m * OUT_CH + c] = v;
}

// ---------------------------------------------------------------------------
extern "C" void kernel_launch(void* const* d_in, const int* in_sizes, int n_in,
                              void* d_out, int out_size, void* d_ws, size_t ws_size,
                              hipStream_t stream) {
  const float* x     = (const float*)d_in[0];
  const float* pos   = (const float*)d_in[1];
  const int*   batch = (const int*)d_in[2];
  const float* W     = (const float*)d_in[3];
  const float* b     = (const float*)d_in[4];
  const float* gamma = (const float*)d_in[5];
  const float* beta  = (const float*)d_in[6];

  char* ws = (char*)d_ws;
  float* h     = (float*)ws;                    // 50000*256*4 = 51,200,000 B
  float* stats = (float*)(ws + 51200000);       // 512 floats
  float* scsh  = (float*)(ws + 51204096);       // 512 floats
  float* cand  = (float*)(ws + 51208192);       // 2*16*8 floats (double-buffered)
  int*   cnt   = (int*)(ws + 51212288);         // grid-barrier counter
  int*   fidx  = (int*)(ws + 51216384);         // 5000 ints
  int*   nn    = (int*)(ws + 51236384);         // 5000*16 ints

  float* out_x     = (float*)d_out;                         // [M, 256]
  float* out_pos   = out_x + (size_t)M_CTR * OUT_CH;        // [M, 3]
  int*   out_batch = (int*)(out_pos + (size_t)M_CTR * 3);   // [M]

  init_ws_kernel<<<1, 512, 0, stream>>>(stats, cnt);
  fps_coop_kernel<<<FPS_NB, FPS_T, 0, stream>>>(pos, cand, cnt, fidx);
  gemm_kernel<<<dim3(N_PTS / 16, OUT_CH / 128), 256, 0, stream>>>(x, W, b, h);
  bn_stats_kernel<<<512, 256, 0, stream>>>(h, stats);
  bn_finalize_kernel<<<1, 256, 0, stream>>>(stats, gamma, beta, scsh);
  bn_relu_kernel<<<N_PTS, 256, 0, stream>>>(h, scsh);
  knn_kernel<<<(M_CTR + 255) / 256, 256, 0, stream>>>(pos, fidx, batch, nn, out_pos, out_batch);
  gather_max_kernel<<<M_CTR, 256, 0, stream>>>(h, nn, out_x);
}